// MemoryBuffer_53317724012937
// MI455X (gfx1250) — compile-verified
//
#include <hip/hip_runtime.h>
#include <math.h>

typedef __attribute__((ext_vector_type(2))) float v2f;
typedef __attribute__((ext_vector_type(8))) float v8f;

#define BATCH 32
#define KDIM  512
#define VDIM  512
#define MSLOT 2048

// ---------------------------------------------------------------------------
// Kernel 1: raw[b][m] = dot(key_mem[b][:, m], x[b]) for m = 0..M-1
// One wave per (b, 16-m tile). WMMA f32 16x16x4, accumulate over f in steps of 4.
// A[i][k] = key_mem[b][f0+k][m0+i]  (i = m-row, k = f)  -> coalesced 64B per half-wave
// B[k][j] = x[b][f0+k] broadcast across all 16 columns j.
// ---------------------------------------------------------------------------
__global__ __launch_bounds__(32)
void mb_scores_kernel(const float* __restrict__ key_mem,
                      const float* __restrict__ x,
                      float* __restrict__ raw) {
    const int b    = blockIdx.x;          // 0..31
    const int mt   = blockIdx.y;          // 0..127
    const int lane = threadIdx.x;         // 0..31
    const int m    = mt * 16 + (lane & 15);
    const int kh   = (lane >> 4) << 1;    // 0 for lanes 0-15, 2 for lanes 16-31

    const float* __restrict__ K  = key_mem + (size_t)b * KDIM * MSLOT;
    const float* __restrict__ xb = x + (size_t)b * KDIM;

    v8f c = {};
    #pragma unroll 8
    for (int f0 = 0; f0 < KDIM; f0 += 4) {
        v2f a;
        a.x = K[(size_t)(f0 + kh)     * MSLOT + m];   // A: K=kh   row
        a.y = K[(size_t)(f0 + kh + 1) * MSLOT + m];   // A: K=kh+1 row
        v2f bv = *(const v2f*)(xb + f0 + kh);         // B: x broadcast pattern
        c = __builtin_amdgcn_wmma_f32_16x16x4_f32(
                /*neg_a=*/false, a, /*neg_b=*/false, bv,
                /*c_mod=*/(short)0, c, /*reuse_a=*/false, /*reuse_b=*/false);
    }
    // D layout: lane 0 VGPR r = D[r][0] (m0+0..7); lane 16 VGPR r = D[r+8][0]
    if ((lane & 15) == 0) {
        const int mbase = mt * 16 + (lane >> 4) * 8;
        float* dst = raw + (size_t)b * MSLOT + mbase;
        #pragma unroll
        for (int r = 0; r < 8; ++r) dst[r] = c[r];
    }
}

// ---------------------------------------------------------------------------
// Kernel 2: per-batch softmax over { s0 = dot(key_in, x) } U { raw[0..M-2] }.
// Writes wshift[b][i] = weight[i+1] for i<M-1, wshift[b][M-1] = 0, w0[b] = weight[0].
// (raw[M-1] corresponds to the dropped last memory slot.)
// ---------------------------------------------------------------------------
__global__ __launch_bounds__(256)
void mb_softmax_kernel(const float* __restrict__ raw,
                       const float* __restrict__ key_in,
                       const float* __restrict__ x,
                       float* __restrict__ wshift,
                       float* __restrict__ w0) {
    const int b = blockIdx.x;
    const int t = threadIdx.x;
    __shared__ float red[256];
    __shared__ float sh_s0, sh_max, sh_sum;

    // s0 = dot(key_in[b], x[b]) over 512 elements
    const float* kb = key_in + (size_t)b * KDIM;
    const float* xb = x      + (size_t)b * KDIM;
    red[t] = kb[t] * xb[t] + kb[t + 256] * xb[t + 256];
    __syncthreads();
    for (int off = 128; off > 0; off >>= 1) {
        if (t < off) red[t] += red[t + off];
        __syncthreads();
    }
    if (t == 0) sh_s0 = red[0];
    __syncthreads();
    const float s0 = sh_s0;

    const float* rb = raw + (size_t)b * MSLOT;

    // max over all 2048 scores (s0 plus raw[0..M-2])
    float lm = (t == 0) ? s0 : -INFINITY;
    for (int i = t; i < MSLOT - 1; i += 256) lm = fmaxf(lm, rb[i]);
    red[t] = lm; __syncthreads();
    for (int off = 128; off > 0; off >>= 1) {
        if (t < off) red[t] = fmaxf(red[t], red[t + off]);
        __syncthreads();
    }
    if (t == 0) sh_max = red[0];
    __syncthreads();
    const float mx = sh_max;

    // sum of exp
    float ls = (t == 0) ? __expf(s0 - mx) : 0.0f;
    for (int i = t; i < MSLOT - 1; i += 256) ls += __expf(rb[i] - mx);
    red[t] = ls; __syncthreads();
    for (int off = 128; off > 0; off >>= 1) {
        if (t < off) red[t] += red[t + off];
        __syncthreads();
    }
    if (t == 0) sh_sum = red[0];
    __syncthreads();
    const float inv = 1.0f / sh_sum;

    float* wb = wshift + (size_t)b * MSLOT;
    for (int i = t; i < MSLOT; i += 256)
        wb[i] = (i < MSLOT - 1) ? __expf(rb[i] - mx) * inv : 0.0f;
    if (t == 0) w0[b] = __expf(s0 - mx) * inv;
}

// ---------------------------------------------------------------------------
// Kernel 3: out[b][v] = value_in[b][v]*w0[b] + sum_m value_mem[b][v][m]*wshift[b][m]
// One wave per (b, 16-v tile). WMMA f32 16x16x4, accumulate over m in steps of 4.
// A[i][k] = value_mem[b][v0+i][m0+k]; B[k][j] = wshift[b][m0+k] broadcast.
// ---------------------------------------------------------------------------
__global__ __launch_bounds__(32)
void mb_out_kernel(const float* __restrict__ value_mem,
                   const float* __restrict__ value_in,
                   const float* __restrict__ wshift,
                   const float* __restrict__ w0,
                   float* __restrict__ out) {
    const int b    = blockIdx.x;          // 0..31
    const int vt   = blockIdx.y;          // 0..31
    const int lane = threadIdx.x;
    const int v    = vt * 16 + (lane & 15);
    const int mh   = (lane >> 4) << 1;    // 0 or 2

    const float* __restrict__ V  = value_mem + ((size_t)b * VDIM + v) * MSLOT;
    const float* __restrict__ wb = wshift + (size_t)b * MSLOT;

    v8f c = {};
    #pragma unroll 8
    for (int m0 = 0; m0 < MSLOT; m0 += 4) {
        v2f a  = *(const v2f*)(V  + m0 + mh);   // A: row v, K=mh..mh+1 (8B aligned)
        v2f bv = *(const v2f*)(wb + m0 + mh);   // B: weight broadcast
        c = __builtin_amdgcn_wmma_f32_16x16x4_f32(
                false, a, false, bv, (short)0, c, false, false);
    }
    if ((lane & 15) == 0) {
        const int vbase = vt * 16 + (lane >> 4) * 8;
        const float wz = w0[b];
        float* dst = out + (size_t)b * VDIM + vbase;
        const float* vin = value_in + (size_t)b * VDIM + vbase;
        #pragma unroll
        for (int r = 0; r < 8; ++r) dst[r] = c[r] + vin[r] * wz;
    }
}

// ---------------------------------------------------------------------------
extern "C" void kernel_launch(void* const* d_in, const int* in_sizes, int n_in,
                              void* d_out, int out_size, void* d_ws, size_t ws_size,
                              hipStream_t stream) {
    const float* key_mem   = (const float*)d_in[0];  // (B, KD, M)
    const float* value_mem = (const float*)d_in[1];  // (B, VD, M)
    const float* x         = (const float*)d_in[2];  // (B, KD)
    const float* key_in    = (const float*)d_in[3];  // (B, KD)
    const float* value_in  = (const float*)d_in[4];  // (B, VD)
    // d_in[5] = II shift matrix: structural, unused.

    float* out = (float*)d_out;                       // (B, VD)

    float* raw    = (float*)d_ws;                     // B*M floats
    float* wshift = raw + (size_t)BATCH * MSLOT;      // B*M floats
    float* w0     = wshift + (size_t)BATCH * MSLOT;   // B floats

    dim3 g1(BATCH, MSLOT / 16);
    mb_scores_kernel<<<g1, 32, 0, stream>>>(key_mem, x, raw);

    mb_softmax_kernel<<<BATCH, 256, 0, stream>>>(raw, key_in, x, wshift, w0);

    dim3 g3(BATCH, VDIM / 16);
    mb_out_kernel<<<g3, 32, 0, stream>>>(value_mem, value_in, wshift, w0, out);
}